// HungarianMatcher_66425964200598
// MI455X (gfx1250) — compile-verified
//
#include <hip/hip_runtime.h>
#include <cstdint>

// Problem shape (fixed by the reference): bs=8, Q=900, C=91, T=200.
// Reference flattens batch: NQ=7200 queries vs NT=1600 targets -> out [7200,1600] f32.
#define NQ       7200
#define NT       1600
#define NC       91
#define QPB      8      // queries per block = one query per wave32
#define THREADS  256    // 8 waves

#define C_CLASS  2.0f
#define C_BBOX   5.0f
#define C_GIOU   2.0f
#define F_ALPHA  0.25f
#define EPSF     1e-8f

// Low 32 bits of a generic LDS pointer == wave-relative LDS byte address
// (flat-to-LDS uses the low 32 address bits; the async LDS destination is an offset from the LDS base).
__device__ __forceinline__ uint32_t lds_addr(const void* p) {
  return (uint32_t)(uintptr_t)p;
}

// CDNA5 async data mover: global -> LDS, tracked by ASYNCcnt.
__device__ __forceinline__ void async_to_lds_b128(uint32_t lds, const void* g) {
  asm volatile("global_load_async_to_lds_b128 %0, %1, off"
               :: "v"(lds), "v"(g) : "memory");
}
__device__ __forceinline__ void async_to_lds_b32(uint32_t lds, const void* g) {
  asm volatile("global_load_async_to_lds_b32 %0, %1, off"
               :: "v"(lds), "v"(g) : "memory");
}
__device__ __forceinline__ void wait_async_zero() {
  asm volatile("s_wait_asynccnt 0" ::: "memory");
}

__global__ __launch_bounds__(THREADS)
void hungarian_cost_kernel(const float* __restrict__ pred_logits,  // [NQ,NC]
                           const float* __restrict__ pred_boxes,   // [NQ,4] cxcywh
                           const int*   __restrict__ tgt_ids,      // [NT]
                           const float* __restrict__ tgt_boxes,    // [NT,4] cxcywh
                           float* __restrict__ out)                // [NQ,NT]
{
  __shared__ __align__(16) float s_tb[NT][4];     // target cxcywh   (25.6 KB)
  __shared__ __align__(16) float s_xyxy[NT][4];   // target xyxy     (25.6 KB)
  __shared__ float s_area[NT];                    // target area     ( 6.4 KB)
  __shared__ int   s_id[NT];                      // target class id ( 6.4 KB)
  __shared__ float s_delta[QPB][NC + 1];          // focal pos-neg per wave, stride 92

  const int tid  = threadIdx.x;
  const int lane = tid & 31;
  const int wave = tid >> 5;
  const int q    = blockIdx.x * QPB + wave;       // NQ % QPB == 0 -> always valid

  // ---- Phase 1: async-stage target boxes (float4) and ids into LDS ----
  for (int i = tid; i < NT; i += THREADS)
    async_to_lds_b128(lds_addr(&s_tb[i][0]), &tgt_boxes[i * 4]);
  for (int i = tid; i < NT; i += THREADS)
    async_to_lds_b32(lds_addr(&s_id[i]), &tgt_ids[i]);
  wait_async_zero();
  __syncthreads();

  // ---- Phase 2a: derive target xyxy + area in LDS ----
  for (int i = tid; i < NT; i += THREADS) {
    float cx = s_tb[i][0], cy = s_tb[i][1], w = s_tb[i][2], h = s_tb[i][3];
    float x0 = cx - 0.5f * w, y0 = cy - 0.5f * h;
    float x1 = cx + 0.5f * w, y1 = cy + 0.5f * h;
    s_xyxy[i][0] = x0; s_xyxy[i][1] = y0; s_xyxy[i][2] = x1; s_xyxy[i][3] = y1;
    s_area[i] = (x1 - x0) * (y1 - y0);
  }

  // ---- Phase 2b: per-query focal table delta[c] = pos[c] - neg[c] ----
  // Queries are partitioned across blocks -> each table computed exactly once.
  for (int c = lane; c < NC; c += 32) {
    float x = pred_logits[(size_t)q * NC + c];
    float p = 1.0f / (1.0f + __expf(-x));
    float pos = F_ALPHA * (1.0f - p) * (1.0f - p) * (-__logf(p + EPSF));
    float neg = (1.0f - F_ALPHA) * p * p * (-__logf(1.0f - p + EPSF));
    s_delta[wave][c] = pos - neg;
  }
  __syncthreads();

  // ---- Query box (uniform across the wave) ----
  const float qcx = pred_boxes[q * 4 + 0];
  const float qcy = pred_boxes[q * 4 + 1];
  const float qw  = pred_boxes[q * 4 + 2];
  const float qh  = pred_boxes[q * 4 + 3];
  const float qx0 = qcx - 0.5f * qw, qy0 = qcy - 0.5f * qh;
  const float qx1 = qcx + 0.5f * qw, qy1 = qcy + 0.5f * qh;
  const float qarea = (qx1 - qx0) * (qy1 - qy0);

  // ---- Main sweep: 32 consecutive targets per iteration (128B coalesced store/wave) ----
  float* __restrict__ orow = out + (size_t)q * NT;
  for (int t0 = 0; t0 < NT; t0 += 32) {
    const int t = t0 + lane;

    const float tcx = s_tb[t][0], tcy = s_tb[t][1], tw = s_tb[t][2], th = s_tb[t][3];
    const float l1 = fabsf(qcx - tcx) + fabsf(qcy - tcy) + fabsf(qw - tw) + fabsf(qh - th);

    const float tx0 = s_xyxy[t][0], ty0 = s_xyxy[t][1];
    const float tx1 = s_xyxy[t][2], ty1 = s_xyxy[t][3];
    const float ta  = s_area[t];

    const float ix0 = fmaxf(qx0, tx0), iy0 = fmaxf(qy0, ty0);
    const float ix1 = fminf(qx1, tx1), iy1 = fminf(qy1, ty1);
    const float iw  = fmaxf(ix1 - ix0, 0.0f), ih = fmaxf(iy1 - iy0, 0.0f);
    const float inter = iw * ih;
    const float uni   = qarea + ta - inter;
    const float iou   = inter / (uni + EPSF);

    const float ex0 = fminf(qx0, tx0), ey0 = fminf(qy0, ty0);
    const float ex1 = fmaxf(qx1, tx1), ey1 = fmaxf(qy1, ty1);
    const float ew  = fmaxf(ex1 - ex0, 0.0f), eh = fmaxf(ey1 - ey0, 0.0f);
    const float earea = ew * eh;
    const float giou  = iou - (earea - uni) / (earea + EPSF);

    const float cls = s_delta[wave][s_id[t]];

    orow[t] = C_BBOX * l1 + C_CLASS * cls + C_GIOU * (-giou);
  }
}

extern "C" void kernel_launch(void* const* d_in, const int* in_sizes, int n_in,
                              void* d_out, int out_size, void* d_ws, size_t ws_size,
                              hipStream_t stream) {
  (void)in_sizes; (void)n_in; (void)out_size; (void)d_ws; (void)ws_size;
  const float* pred_logits = (const float*)d_in[0];   // [8,900,91]  f32
  const float* pred_boxes  = (const float*)d_in[1];   // [8,900,4]   f32
  const int*   tgt_ids     = (const int*)  d_in[2];   // [8,200]     int
  const float* tgt_boxes   = (const float*)d_in[3];   // [8,200,4]   f32
  float* out = (float*)d_out;                         // [8,900,1600] f32

  dim3 grid(NQ / QPB);   // 900 blocks, 8 wave32s each
  hungarian_cost_kernel<<<grid, THREADS, 0, stream>>>(
      pred_logits, pred_boxes, tgt_ids, tgt_boxes, out);
}